// S6_9363028705795
// MI455X (gfx1250) — compile-verified
//
#include <hip/hip_runtime.h>
#include <math.h>

// ---------------------------------------------------------------- constants
#define BATCH   2
#define SEQLEN  2048
#define D_INNER 2048
#define D_STATE 16
#define DT_RANK 64
#define KPROJ   (DT_RANK + 2 * D_STATE)   // 96

typedef float v2f __attribute__((ext_vector_type(2)));
typedef float v8f __attribute__((ext_vector_type(8)));

// =========================================================================
// Kernel 1: x_dbl[b,l,k] = sum_d u[b,d,l] * x_proj_w[k,d]
// GEMM M=B*L=4096 (m=(b,l)), K=D_INNER=2048, N=KPROJ=96, f32 WMMA 16x16x4.
// 4 interleaved accumulators break the WMMA->WMMA RAW chain (4x shorter
// serial dependency; only ~1.5 waves/SIMD available to hide it otherwise).
// =========================================================================
__global__ __launch_bounds__(256) void xdbl_gemm_wmma(
    const float* __restrict__ u,        // (B, D, L)
    const float* __restrict__ xpw,      // (KPROJ, D)
    float* __restrict__ xdbl)           // (B, L, KPROJ)
{
    const int lane = threadIdx.x & 31;
    const int wave = threadIdx.x >> 5;
    const int tile = blockIdx.x * 8 + wave;         // 1536 tiles total
    const int nt = tile % (KPROJ / 16);             // 0..5
    const int mt = tile / (KPROJ / 16);             // 0..255
    const int m0 = mt * 16, n0 = nt * 16;

    const int half = lane & 15;                     // M row (A) / N col (B,C)
    const int hi   = lane >> 4;                     // 0: K pair {0,1}, 1: {2,3}
    const int kk   = hi * 2;

    const int b = m0 >> 11;                         // m0 / SEQLEN (tiles never cross batch)
    const int l = (m0 & (SEQLEN - 1)) + half;

    // A[m][k] = u[b, k, l]  -> stride SEQLEN in k
    const float* pa = u + (size_t)b * D_INNER * SEQLEN + (size_t)kk * SEQLEN + l;
    // B[k][n] = xpw[n*D + k]
    const float* pb = xpw + (size_t)(n0 + half) * D_INNER + kk;

    v8f c0 = {}, c1 = {}, c2 = {}, c3 = {};
    #pragma unroll 2
    for (int k = 0; k < D_INNER; k += 16) {
        v2f a, bm;
        a.x  = pa[0 * SEQLEN];  a.y  = pa[1 * SEQLEN];
        bm.x = pb[0];           bm.y = pb[1];
        c0 = __builtin_amdgcn_wmma_f32_16x16x4_f32(false, a, false, bm, (short)0, c0, false, false);

        a.x  = pa[4 * SEQLEN];  a.y  = pa[5 * SEQLEN];
        bm.x = pb[4];           bm.y = pb[5];
        c1 = __builtin_amdgcn_wmma_f32_16x16x4_f32(false, a, false, bm, (short)0, c1, false, false);

        a.x  = pa[8 * SEQLEN];  a.y  = pa[9 * SEQLEN];
        bm.x = pb[8];           bm.y = pb[9];
        c2 = __builtin_amdgcn_wmma_f32_16x16x4_f32(false, a, false, bm, (short)0, c2, false, false);

        a.x  = pa[12 * SEQLEN]; a.y  = pa[13 * SEQLEN];
        bm.x = pb[12];          bm.y = pb[13];
        c3 = __builtin_amdgcn_wmma_f32_16x16x4_f32(false, a, false, bm, (short)0, c3, false, false);

        pa += 16 * SEQLEN;
        pb += 16;
    }
    v8f c = (c0 + c1) + (c2 + c3);

    // C/D layout: VGPR r -> M = r + hi*8, N = lane&15
    float* pc = xdbl + (size_t)(m0 + hi * 8) * KPROJ + (n0 + half);
    #pragma unroll
    for (int r = 0; r < 8; ++r)
        pc[(size_t)r * KPROJ] = c[r];
}

// =========================================================================
// Kernel 2: delta[b,d,l] = softplus( sum_r x_dbl[b,l,r] * dt_proj_w[d,r] + b[d] )
// GEMM M=B*L=4096, K=DT_RANK=64, N=D_INNER=2048, f32 WMMA 16x16x4 with
// 2 interleaved accumulators, fused bias + stable softplus, stored in
// (b,d,l) layout so the scan reads it contiguously per channel.
// =========================================================================
__global__ __launch_bounds__(256) void delta_gemm_wmma(
    const float* __restrict__ xdbl,     // (B, L, KPROJ)
    const float* __restrict__ dtw,      // (D, DT_RANK)
    const float* __restrict__ dtb,      // (D)
    float* __restrict__ delta)          // (B, D, L)
{
    const int lane = threadIdx.x & 31;
    const int wave = threadIdx.x >> 5;
    const int tile = blockIdx.x * 8 + wave;         // 32768 tiles total
    const int nt = tile & 127;                      // D_INNER/16 = 128
    const int mt = tile >> 7;                       // 0..255
    const int m0 = mt * 16, n0 = nt * 16;

    const int half = lane & 15;
    const int hi   = lane >> 4;
    const int kk   = hi * 2;

    // A[m][r] = xdbl[m*KPROJ + r]   (first DT_RANK columns of x_dbl)
    const float* pa = xdbl + (size_t)(m0 + half) * KPROJ + kk;
    // B[r][d] = dtw[d*DT_RANK + r]
    const float* pb = dtw + (size_t)(n0 + half) * DT_RANK + kk;

    v8f c0 = {}, c1 = {};
    #pragma unroll
    for (int k = 0; k < DT_RANK; k += 8) {
        v2f a, bm;
        a.x  = pa[0];  a.y  = pa[1];
        bm.x = pb[0];  bm.y = pb[1];
        c0 = __builtin_amdgcn_wmma_f32_16x16x4_f32(false, a, false, bm, (short)0, c0, false, false);

        a.x  = pa[4];  a.y  = pa[5];
        bm.x = pb[4];  bm.y = pb[5];
        c1 = __builtin_amdgcn_wmma_f32_16x16x4_f32(false, a, false, bm, (short)0, c1, false, false);

        pa += 8;
        pb += 8;
    }
    v8f c = c0 + c1;

    const int   d    = n0 + half;                   // N col of this lane
    const float bias = dtb[d];
    const int   b    = m0 >> 11;
    const int   lb   = (m0 & (SEQLEN - 1)) + hi * 8;
    float* pd = delta + ((size_t)b * D_INNER + d) * SEQLEN + lb;

    #pragma unroll
    for (int r = 0; r < 8; ++r) {
        float z  = c[r] + bias;
        // softplus(z) = max(z,0) + log1p(exp(-|z|))  (numerically stable)
        float sp = fmaxf(z, 0.0f) + log1pf(__expf(-fabsf(z)));
        pd[r] = sp;
    }
}

// =========================================================================
// Kernel 3: selective scan.  16 states of one (b,d) channel live on 16 lanes;
// two channels per wave32 -> 2048 waves of occupancy (2 per SIMD).
//   x_n  = exp(delta*A_n) * x_n + (delta*u) * B_n
//   y    = sum_n x_n * C_n   (4-step shfl_xor butterfly within 16-lane group)
// Unroll x4: loads of the next iterations are independent of the x chain,
// so the compiler can hoist them ahead of the dependent mul->exp->fma chain.
// =========================================================================
__global__ __launch_bounds__(256) void s6_scan(
    const float* __restrict__ u,        // (B, D, L)
    const float* __restrict__ xdbl,     // (B, L, KPROJ): [.. r | B 16 | C 16]
    const float* __restrict__ delta,    // (B, D, L)
    const float* __restrict__ A_log,    // (D, N)
    float* __restrict__ out)            // (B, D, L)
{
    const int lane = threadIdx.x & 31;
    const int gwave = (blockIdx.x * blockDim.x + threadIdx.x) >> 5;
    const int g = lane >> 4;                         // sub-channel within wave
    const int n = lane & 15;                         // state index
    const int ch = gwave * 2 + g;                    // 0..B*D-1, ch == b*D + d
    const int b  = ch >> 11;                         // ch / D_INNER
    const int d  = ch & (D_INNER - 1);

    const float An = -__expf(A_log[(size_t)d * D_STATE + n]);

    const float* pdl = delta + (size_t)ch * SEQLEN;
    const float* pu  = u     + (size_t)ch * SEQLEN;
    const float* pbc = xdbl  + (size_t)b * SEQLEN * KPROJ + DT_RANK + n;
    float*       po  = out   + (size_t)ch * SEQLEN;

    float x = 0.0f;
    #pragma unroll 4
    for (int l = 0; l < SEQLEN; ++l) {
        if ((l & 15) == 0) {                         // pull next cachelines
            __builtin_prefetch(pdl + l + 64, 0, 0);
            __builtin_prefetch(pu  + l + 64, 0, 0);
        }
        float dl = pdl[l];
        float uv = pu[l];
        float Bn = pbc[(size_t)l * KPROJ];
        float Cn = pbc[(size_t)l * KPROJ + D_STATE];

        float dA = __expf(dl * An);
        x = dA * x + (dl * uv) * Bn;

        float p = x * Cn;
        p += __shfl_xor(p, 1);
        p += __shfl_xor(p, 2);
        p += __shfl_xor(p, 4);
        p += __shfl_xor(p, 8);
        if (n == 0) po[l] = p;
    }
}

// =========================================================================
extern "C" void kernel_launch(void* const* d_in, const int* in_sizes, int n_in,
                              void* d_out, int out_size, void* d_ws, size_t ws_size,
                              hipStream_t stream)
{
    const float* u     = (const float*)d_in[0];   // (B, D, L)
    const float* xpw   = (const float*)d_in[1];   // (KPROJ, D)
    const float* dtw   = (const float*)d_in[2];   // (D, DT_RANK)
    const float* dtb   = (const float*)d_in[3];   // (D)
    const float* A_log = (const float*)d_in[4];   // (D, N)
    float*       out   = (float*)d_out;           // (B, D, L)

    float* xdbl  = (float*)d_ws;                                  // B*L*KPROJ  (1.5 MB)
    float* delta = xdbl + (size_t)BATCH * SEQLEN * KPROJ;         // B*D*L      (32 MB)

    // GEMM1: 4096/16 * 96/16 = 1536 tiles, 8 waves/block -> 192 blocks
    xdbl_gemm_wmma<<<192, 256, 0, stream>>>(u, xpw, xdbl);

    // GEMM2: 4096/16 * 2048/16 = 32768 tiles, 8 waves/block -> 4096 blocks
    delta_gemm_wmma<<<4096, 256, 0, stream>>>(xdbl, dtw, dtb, delta);

    // Scan: B*D/2 = 2048 waves, 8 waves/block -> 256 blocks
    s6_scan<<<256, 256, 0, stream>>>(u, xdbl, delta, A_log, out);
}